// MultiHeadNeuralCA_22660247453837
// MI455X (gfx1250) — compile-verified
//
#include <hip/hip_runtime.h>
#include <math.h>

// ---------------- problem constants ----------------
#define BB 2
#define HH 8
#define SS 2048
#define DD 64
#define II 512
#define KK 3
#define DLAT 384
#define DSTIM 128

#define NBH   (BB*HH)           // 16
#define NBHS  (BB*HH*SS)        // 32768
#define NPROJ (BB*HH*SS*DD)     // 2097152

// ---------------- workspace layout (float offsets) ----------------
#define WS_P    ((size_t)0)
#define WS_INT  (WS_P   + (size_t)NPROJ)
#define WS_V    (WS_INT + (size_t)NPROJ)
#define WS_CONV (WS_V   + (size_t)NPROJ)
#define WS_GROW (WS_CONV+ (size_t)NPROJ)
#define WS_ROW  (WS_GROW+ (size_t)NBHS)
#define WS_ISUM (WS_ROW + (size_t)NBHS)
#define WS_VSUM (WS_ISUM+ (size_t)(NBH*DD))
#define WS_VSSM (WS_VSUM+ (size_t)(NBH*DD))
#define WS_OPCS (WS_VSSM+ (size_t)(NBH*DD))
#define WS_SCAL (WS_OPCS+ (size_t)(HH*DD))
// scalars: 0=inputMass 1=outputMass 2=entropySum 3=smSum 4=smSumSq

// ---------------- output layout (float offsets in d_out) ----------------
#define OUT_MAIN ((size_t)0)
#define OUT_CORR ((size_t)NPROJ)
#define OUT_SCAL ((size_t)(2*NPROJ))

typedef __attribute__((ext_vector_type(2))) float v2f;
typedef __attribute__((ext_vector_type(8))) float v8f;

// fp32 WMMA: D(16x16,f32) = A(16x4,f32) x B(4x16,f32) + C
__device__ __forceinline__ v8f wmma4(v2f a, v2f b, v8f c) {
  return __builtin_amdgcn_wmma_f32_16x16x4_f32(false, a, false, b, (short)0, c,
                                               false, false);
}

// ---------------- init: zero the accumulator scalars ----------------
__global__ void k_init(float* scal) {
  if (threadIdx.x < 5) scal[threadIdx.x] = 0.0f;
}

// ---------------- input mass: sum(latent)+sum(stimulus) ----------------
__global__ void k_input_mass(const float* __restrict__ lat,
                             const float* __restrict__ stim, float* scal) {
  const size_t NL = (size_t)BB * SS * DLAT;
  const size_t NT = NL + (size_t)BB * SS * DSTIM;
  float acc = 0.0f;
  for (size_t i = (size_t)blockIdx.x * blockDim.x + threadIdx.x; i < NT;
       i += (size_t)gridDim.x * blockDim.x)
    acc += (i < NL) ? lat[i] : stim[i - NL];
  __shared__ float red[256];
  red[threadIdx.x] = acc;
  __syncthreads();
  for (int st = 128; st > 0; st >>= 1) {
    if ((int)threadIdx.x < st) red[threadIdx.x] += red[threadIdx.x + st];
    __syncthreads();
  }
  if (threadIdx.x == 0) atomicAdd(&scal[0], red[0]);
}

// ---------------- variance stats of spatial_modulation weight ----------------
__global__ void k_sm_var(const float* __restrict__ sm, float* scal) {
  const size_t N = (size_t)HH * DD * DD;
  float s = 0.0f, sq = 0.0f;
  for (size_t i = (size_t)blockIdx.x * blockDim.x + threadIdx.x; i < N;
       i += (size_t)gridDim.x * blockDim.x) {
    float v = sm[i];
    s += v; sq += v * v;
  }
  __shared__ float r0[256], r1[256];
  r0[threadIdx.x] = s; r1[threadIdx.x] = sq;
  __syncthreads();
  for (int st = 128; st > 0; st >>= 1) {
    if ((int)threadIdx.x < st) {
      r0[threadIdx.x] += r0[threadIdx.x + st];
      r1[threadIdx.x] += r1[threadIdx.x + st];
    }
    __syncthreads();
  }
  if (threadIdx.x == 0) { atomicAdd(&scal[3], r0[0]); atomicAdd(&scal[4], r1[0]); }
}

// ---------------- projections: P/INT/V[bh,s,d] = x[b,s,:] @ W[h,:,:] ----------------
// block = 128 (4 waves; wave w -> d-tile w). grid = (NBH * S/16, 3 weights)
__global__ void k_proj_wmma(const float* __restrict__ lat,
                            const float* __restrict__ stim,
                            const float* __restrict__ Wp,
                            const float* __restrict__ Wi,
                            const float* __restrict__ Wv,
                            float* __restrict__ ws) {
  const int lane = threadIdx.x & 31;
  const int wave = threadIdx.x >> 5;
  const int st = blockIdx.x % (SS / 16);
  const int bh = blockIdx.x / (SS / 16);
  const int h = bh % HH, b = bh / HH;
  const int wsel = blockIdx.y;
  const float* W = (wsel == 0) ? Wp : (wsel == 1) ? Wi : Wv;
  float* out = ws + ((wsel == 0) ? WS_P : (wsel == 1) ? WS_INT : WS_V);

  const int s0 = st * 16;
  const int mrow = lane & 15;          // A-matrix M row for this lane
  const int koff = (lane >> 4) * 2;    // A-matrix K base for this lane half
  const int d0 = wave * 16;
  const int n = lane & 15;             // B/C column
  const int chalf = (lane >> 4) * 8;   // C row offset for this lane half

  const float* latRow = lat + ((size_t)b * SS + s0 + mrow) * DLAT;
  const float* stiRow = stim + ((size_t)b * SS + s0 + mrow) * DSTIM;
  const float* Wh = W + (size_t)h * II * DD;

  v8f acc = {};
  for (int kb = 0; kb < II; kb += 4) {
    const int c = kb + koff;
    v2f a, bm;
    if (c < DLAT) { a.x = latRow[c]; a.y = latRow[c + 1]; }
    else          { a.x = stiRow[c - DLAT]; a.y = stiRow[c - DLAT + 1]; }
    bm.x = Wh[(size_t)c * DD + d0 + n];
    bm.y = Wh[(size_t)(c + 1) * DD + d0 + n];
    acc = wmma4(a, bm, acc);
  }
  float* ob = out + ((size_t)bh * SS + s0) * DD + d0;
#pragma unroll
  for (int r = 0; r < 8; ++r) ob[(size_t)(r + chalf) * DD + n] = acc[r];
}

// ---------------- correlation = INT * V (elementwise) ----------------
__global__ void k_corr(const float* __restrict__ ws, float* __restrict__ out) {
  const float* Ip = ws + WS_INT;
  const float* Vp = ws + WS_V;
  for (size_t i = (size_t)blockIdx.x * blockDim.x + threadIdx.x; i < (size_t)NPROJ;
       i += (size_t)gridDim.x * blockDim.x)
    out[OUT_CORR + i] = Ip[i] * Vp[i];
}

// ---------------- grouped conv1d: conv[bh,s,d] = sum_{i,k} P[bh,s+k-1,i]*Ker[h,d,i,k]
// block = 128 (4 waves -> d-tiles). grid = NBH * S/16
__global__ void k_conv_wmma(const float* __restrict__ ws_p,
                            const float* __restrict__ ker,
                            float* __restrict__ conv) {
  const int lane = threadIdx.x & 31;
  const int wave = threadIdx.x >> 5;
  const int st = blockIdx.x % (SS / 16);
  const int bh = blockIdx.x / (SS / 16);
  const int h = bh % HH;
  const int s0 = st * 16;
  const int mrow = lane & 15;
  const int koff = (lane >> 4) * 2;
  const int d0 = wave * 16;
  const int n = lane & 15;
  const int chalf = (lane >> 4) * 8;

  const float* kerdn = ker + (((size_t)h * DD + d0 + n) * DD) * KK;  // [i][k]
  v8f acc = {};
  for (int kk = 0; kk < KK; ++kk) {
    const int srow = s0 + mrow + kk - 1;
    const bool ok = (srow >= 0) && (srow < SS);
    const float* prow = ws_p + ((size_t)bh * SS + (ok ? srow : 0)) * DD;
    for (int ib = 0; ib < DD; ib += 4) {
      const int c = ib + koff;
      v2f a, bm;
      a.x = ok ? prow[c] : 0.0f;
      a.y = ok ? prow[c + 1] : 0.0f;
      bm.x = kerdn[(size_t)c * KK + kk];
      bm.y = kerdn[(size_t)(c + 1) * KK + kk];
      acc = wmma4(a, bm, acc);
    }
  }
  float* ob = conv + ((size_t)bh * SS + s0) * DD + d0;
#pragma unroll
  for (int r = 0; r < 8; ++r) ob[(size_t)(r + chalf) * DD + n] = acc[r];
}

// ---------------- Isum/Vsum over sequence ----------------
__global__ void k_reduce_iv(const float* __restrict__ ws, float* __restrict__ wsw) {
  const int bh = blockIdx.x, d = threadIdx.x;  // 64 threads
  const float* ib = ws + WS_INT + (size_t)bh * SS * DD + d;
  const float* vb = ws + WS_V + (size_t)bh * SS * DD + d;
  float is = 0.0f, vs = 0.0f;
  for (int s = 0; s < SS; ++s) { is += ib[(size_t)s * DD]; vs += vb[(size_t)s * DD]; }
  wsw[WS_ISUM + (size_t)bh * DD + d] = is;
  wsw[WS_VSUM + (size_t)bh * DD + d] = vs;
}

// ---------------- growth[q] and rowsum[q] (one wave per q) ----------------
__global__ void k_growth_row(const float* __restrict__ ws,
                             const float* __restrict__ gc,
                             const float* __restrict__ gw,
                             float* __restrict__ wsw) {
  const int lane = threadIdx.x & 31;
  const int wv = threadIdx.x >> 5;
  const size_t q = (size_t)blockIdx.x * 8 + wv;  // global (b,h,s)
  const int bh = (int)(q / SS);
  const int h = bh % HH;
  const float* irow = ws + WS_INT + q * DD;
  const float* crow = ws + WS_CONV + q * DD;
  const float* isum = ws + WS_ISUM + (size_t)bh * DD;
  float pot = irow[lane] + irow[lane + 32];
  float sr = crow[lane] * isum[lane] + crow[lane + 32] * isum[lane + 32];
  for (int off = 16; off > 0; off >>= 1) {
    pot += __shfl_down(pot, off, 32);
    sr += __shfl_down(sr, off, 32);
  }
  if (lane == 0) {
    pot *= (1.0f / DD);
    const float z = (pot - gc[h]) / (gw[h] + 1e-10f);
    const float g = expf(-0.5f * z * z);
    wsw[WS_GROW + q] = g;
    wsw[WS_ROW + q] = g * (sr * 0.125f);  // growth * sum_k scores
  }
}

// ---------------- entropy over the implicit SxS pattern (WMMA tiles) ----------------
// grid = (NBH, S/16 q-tiles), block = 256 (8 waves loop over k-tiles)
__global__ void k_entropy_wmma(const float* __restrict__ ws, float* scal) {
  const int lane = threadIdx.x & 31;
  const int wv = threadIdx.x >> 5;
  const int bh = blockIdx.x;
  const int q0 = blockIdx.y * 16;
  const int mrow = lane & 15;
  const int koff = (lane >> 4) * 2;
  const int n = lane & 15;
  const int chalf = (lane >> 4) * 8;

  // per-lane row params for its 8 C rows
  float gq[8], rin[8];
#pragma unroll
  for (int r = 0; r < 8; ++r) {
    const size_t q = (size_t)bh * SS + q0 + r + chalf;
    gq[r] = ws[WS_GROW + q];
    rin[r] = 1.0f / (ws[WS_ROW + q] + 1e-10f);
  }
  // hoist the A tile (conv rows) — reused for every k-tile
  const float* crow = ws + WS_CONV + ((size_t)bh * SS + q0 + mrow) * DD;
  v2f av[16];
#pragma unroll
  for (int db = 0; db < 16; ++db) {
    const int c = db * 4 + koff;
    av[db].x = crow[c]; av[db].y = crow[c + 1];
  }

  float epart = 0.0f;
  for (int kt = wv; kt < SS / 16; kt += 8) {
    const float* ib = ws + WS_INT + ((size_t)bh * SS + kt * 16 + n) * DD;
    v8f acc = {};
#pragma unroll
    for (int db = 0; db < 16; ++db) {
      const int c = db * 4 + koff;
      v2f bm; bm.x = ib[c]; bm.y = ib[c + 1];
      acc = wmma4(av[db], bm, acc);
    }
#pragma unroll
    for (int r = 0; r < 8; ++r) {
      const float p = acc[r] * 0.125f * gq[r];  // ca_pattern element
      const float pn = p * rin[r];
      epart -= pn * logf(pn + 1e-10f);
    }
  }
  __shared__ float red[256];
  red[threadIdx.x] = epart;
  __syncthreads();
  for (int st = 128; st > 0; st >>= 1) {
    if ((int)threadIdx.x < st) red[threadIdx.x] += red[threadIdx.x + st];
    __syncthreads();
  }
  if (threadIdx.x == 0) atomicAdd(&scal[2], red[0]);
}

// ---------------- VSsm[bh,o] = sum_d Vsum[bh,d]*SM[h,d,o] ----------------
__global__ void k_vssm(const float* __restrict__ ws, const float* __restrict__ sm,
                       float* __restrict__ wsw) {
  const int bh = blockIdx.x, h = bh % HH, o = threadIdx.x;
  const float* vs = ws + WS_VSUM + (size_t)bh * DD;
  float a = 0.0f;
  for (int d = 0; d < DD; ++d) a += vs[d] * sm[((size_t)h * DD + d) * DD + o];
  wsw[WS_VSSM + (size_t)bh * DD + o] = a;
}

// ---------------- OPcolsum[h,d] = sum_o OP[h,d,o] ----------------
__global__ void k_opcolsum(const float* __restrict__ op, float* __restrict__ wsw) {
  const int h = blockIdx.x, d = threadIdx.x;
  const float* r = op + ((size_t)h * DD + d) * DD;
  float a = 0.0f;
  for (int o = 0; o < DD; ++o) a += r[o];
  wsw[WS_OPCS + (size_t)h * DD + d] = a;
}

// ---------------- analytic unscaled output mass (one wave per q) ----------------
__global__ void k_out_mass(const float* __restrict__ ws, float* scal) {
  const int lane = threadIdx.x & 31;
  const int wv = threadIdx.x >> 5;
  const size_t q = (size_t)blockIdx.x * 8 + wv;
  const int bh = (int)(q / SS);
  const int h = bh % HH;
  const float r = ws[WS_ROW + q];
  const float* vs = ws + WS_VSUM + (size_t)bh * DD;
  const float* vm = ws + WS_VSSM + (size_t)bh * DD;
  const float* oc = ws + WS_OPCS + (size_t)h * DD;
  float part = r * vs[lane] * tanhf(r * vm[lane]) * oc[lane] +
               r * vs[lane + 32] * tanhf(r * vm[lane + 32]) * oc[lane + 32];
  for (int off = 16; off > 0; off >>= 1) part += __shfl_down(part, off, 32);
  __shared__ float red[8];
  if (lane == 0) red[wv] = part;
  __syncthreads();
  if (threadIdx.x == 0) {
    float s = 0.0f;
    for (int i = 0; i < 8; ++i) s += red[i];
    atomicAdd(&scal[1], s);
  }
}

// ---------------- final output projection (WMMA) with mass conservation ----------------
// block = 128 (4 waves -> o-tiles). grid = NBH * S/16
__global__ void k_final_wmma(const float* __restrict__ ws,
                             const float* __restrict__ op,
                             float* __restrict__ out) {
  const int lane = threadIdx.x & 31;
  const int wave = threadIdx.x >> 5;
  const int st = blockIdx.x % (SS / 16);
  const int bh = blockIdx.x / (SS / 16);
  const int h = bh % HH;
  const int s0 = st * 16;
  const int koff = (lane >> 4) * 2;
  const int o0 = wave * 16;
  const int n = lane & 15;
  const int chalf = (lane >> 4) * 8;

  const float im = ws[WS_SCAL + 0], om = ws[WS_SCAL + 1];
  const float scale = (fabsf(om) > 1e-10f) ? (im / om) : 1.0f;

  const int q = s0 + (lane & 15);
  const float r = ws[WS_ROW + (size_t)bh * SS + q];
  const float* vs = ws + WS_VSUM + (size_t)bh * DD;
  const float* vm = ws + WS_VSSM + (size_t)bh * DD;
  const float* oph = op + (size_t)h * DD * DD;

  v8f acc = {};
  for (int db = 0; db < DD; db += 4) {
    const int c = db + koff;
    v2f a, bm;
    a.x = r * vs[c] * tanhf(r * vm[c]);
    a.y = r * vs[c + 1] * tanhf(r * vm[c + 1]);
    bm.x = oph[(size_t)c * DD + o0 + n];
    bm.y = oph[(size_t)(c + 1) * DD + o0 + n];
    acc = wmma4(a, bm, acc);
  }
  float* ob = out + OUT_MAIN + ((size_t)bh * SS + s0) * DD + o0;
#pragma unroll
  for (int rr = 0; rr < 8; ++rr)
    ob[(size_t)(rr + chalf) * DD + n] = acc[rr] * scale;
}

// ---------------- finalize the three scalar metrics ----------------
__global__ void k_finalize(const float* __restrict__ ws, float* __restrict__ out) {
  if (threadIdx.x != 0 || blockIdx.x != 0) return;
  const float im = ws[WS_SCAL + 0], om = ws[WS_SCAL + 1];
  const float ent = ws[WS_SCAL + 2], ssum = ws[WS_SCAL + 3], ssq = ws[WS_SCAL + 4];
  const float scale = (fabsf(om) > 1e-10f) ? (im / om) : 1.0f;
  const float mass_out = om * scale;
  out[OUT_SCAL + 0] = 1.0f - fabsf(mass_out - im) / (fabsf(im) + 1e-10f);
  const float N = (float)(HH * DD * DD);
  const float mean = ssum / N;
  const float var = ssq / N - mean * mean;
  out[OUT_SCAL + 1] = 1.0f / (1.0f + expf(-var));           // sigmoid(var)
  const float em = ent / (float)NBHS;
  out[OUT_SCAL + 2] = 1.0f / (1.0f + expf(em));             // sigmoid(-mean entropy)
}

// ---------------- host launcher ----------------
extern "C" void kernel_launch(void* const* d_in, const int* in_sizes, int n_in,
                              void* d_out, int out_size, void* d_ws, size_t ws_size,
                              hipStream_t stream) {
  (void)in_sizes; (void)n_in; (void)out_size; (void)ws_size;
  const float* lat = (const float*)d_in[0];
  const float* stim = (const float*)d_in[1];
  const float* Wp = (const float*)d_in[2];
  const float* Wi = (const float*)d_in[3];
  const float* Wv = (const float*)d_in[4];
  const float* ker = (const float*)d_in[5];
  const float* sm = (const float*)d_in[6];
  const float* op = (const float*)d_in[7];
  const float* gc = (const float*)d_in[8];
  const float* gw = (const float*)d_in[9];
  float* out = (float*)d_out;
  float* ws = (float*)d_ws;
  float* scal = ws + WS_SCAL;

  k_init<<<1, 32, 0, stream>>>(scal);
  k_input_mass<<<512, 256, 0, stream>>>(lat, stim, scal);
  k_sm_var<<<32, 256, 0, stream>>>(sm, scal);

  // projections: 2048 blocks x 3 weights, 4 waves each
  k_proj_wmma<<<dim3(NBH * (SS / 16), 3), 128, 0, stream>>>(lat, stim, Wp, Wi, Wv, ws);
  k_corr<<<1024, 256, 0, stream>>>(ws, out);
  k_conv_wmma<<<NBH * (SS / 16), 128, 0, stream>>>(ws + WS_P, ker, ws + WS_CONV);
  k_reduce_iv<<<NBH, DD, 0, stream>>>(ws, ws);
  k_growth_row<<<NBHS / 8, 256, 0, stream>>>(ws, gc, gw, ws);
  k_entropy_wmma<<<dim3(NBH, SS / 16), 256, 0, stream>>>(ws, scal);
  k_vssm<<<NBH, DD, 0, stream>>>(ws, sm, ws);
  k_opcolsum<<<HH, DD, 0, stream>>>(op, ws);
  k_out_mass<<<NBHS / 8, 256, 0, stream>>>(ws, scal);
  k_final_wmma<<<NBH * (SS / 16), 128, 0, stream>>>(ws, op, out);
  k_finalize<<<1, 32, 0, stream>>>(ws, out);
}